// ConvLRU_11398843203895
// MI455X (gfx1250) — compile-verified
//
#include <hip/hip_runtime.h>
#include <math.h>

// Problem constants (B,L,E,C,H,W) = (4,32,64,96,32,32)
#define BB 4
#define LL 32
#define EE 64
#define CC 96
#define HH 32
#define WW 32
#define SS 32   // anti-diagonal index (h+w)%32
#define KK 32   // frequency index (diagonal of FFT2)
#define BL (BB*LL)

typedef float v8f __attribute__((ext_vector_type(8)));
typedef float v2f __attribute__((ext_vector_type(2)));

// ---------------------------------------------------------------------------
// WMMA f32 16x16x4 fragment helpers (wave32, EXEC all-ones at call sites).
// A 16x4 (MxK): lanes 0-15 -> M=lane, K={0,1}; lanes 16-31 -> M=lane-16, K={2,3}
// B 4x16 (KxN): lanes 0-15 -> N=lane, K={0,1}; lanes 16-31 -> N=lane-16, K={2,3}
// C/D 16x16  : VGPR v -> row v (lanes 0-15) / row v+8 (lanes 16-31), col=lane&15
// ---------------------------------------------------------------------------
__device__ __forceinline__ v2f frag_a(const float* A, int lda) {
  int lane = threadIdx.x & 31;
  int m = lane & 15;
  int k = (lane >> 4) << 1;
  v2f a;
  a.x = A[m * lda + k];
  a.y = A[m * lda + k + 1];
  return a;
}

__device__ __forceinline__ v2f frag_b(const float* Bm, int ldb) {
  int lane = threadIdx.x & 31;
  int n = lane & 15;
  int k = (lane >> 4) << 1;
  v2f b;
  b.x = Bm[k * ldb + n];
  b.y = Bm[(k + 1) * ldb + n];
  return b;
}

__device__ __forceinline__ v8f wmma4(v2f a, v2f b, v8f acc) {
  // 8 args: (neg_a, A, neg_b, B, c_mod, C, reuse_a, reuse_b)
  return __builtin_amdgcn_wmma_f32_16x16x4_f32(false, a, false, b, (short)0, acc,
                                               false, false);
}

// acc += A(16xK) * B(Kx16), row-major A (lda), row-major B (ldb); negB subtracts
__device__ __forceinline__ v8f mm_tile(const float* A, int lda, const float* Bm,
                                       int ldb, int Ktot, v8f acc, bool negB) {
  for (int k = 0; k < Ktot; k += 4) {
    v2f a = frag_a(A + k, lda);
    v2f b = frag_b(Bm + k * ldb, ldb);
    if (negB) { b.x = -b.x; b.y = -b.y; }
    acc = wmma4(a, b, acc);
  }
  return acc;
}

// ---------------------------------------------------------------------------
// K1: y[bl,e,s] = sum_h x[bl,e,h,(s-h)&31]   (fully coalesced rotated rows)
// ---------------------------------------------------------------------------
__global__ void k1_diag(const float* __restrict__ x, float* __restrict__ y) {
  int t = blockIdx.x * 256 + threadIdx.x;      // BL*E*S = 262144 threads
  int s = t & 31;
  const float* xp = x + (t >> 5) * 1024;       // (bl*E + e) * H*W
  float acc = 0.f;
#pragma unroll
  for (int h = 0; h < 32; ++h)
    acc += xp[h * 32 + ((s - h) & 31)];
  y[t] = acc;                                  // t == bl*2048 + e*32 + s
}

// ---------------------------------------------------------------------------
// K2 (per (b,l) block): g = Wb*y + 32*bb ; D = g*F (DFT32) ; v = gamma⊙(Wp1*D+bp1)
// ---------------------------------------------------------------------------
__global__ void __launch_bounds__(256)
k2_front(const float* __restrict__ y,
         const float* __restrict__ Wbr, const float* __restrict__ Wbi,
         const float* __restrict__ bbr, const float* __restrict__ bbi,
         const float* __restrict__ Wp1r, const float* __restrict__ Wp1i,
         const float* __restrict__ bp1r, const float* __restrict__ bp1i,
         const float* __restrict__ plog,            // (3C,32): nu/theta/gamma logs
         float* __restrict__ vr, float* __restrict__ vi) {
  __shared__ float lds[16384];                      // 64 KB
  float* ylds = lds;                                // 2048  (E x S)
  float* Fr   = lds + 2048;                         // 1024  F[s][k]=cos
  float* Fi   = lds + 3072;                         // 1024  F[s][k]=-sin
  float* gr   = lds + 4096;                         // 3072  (C x S)
  float* gi   = lds + 7168;                         // 3072
  float* Dr   = lds + 10240;                        // 3072  (C x K)
  float* Di   = lds + 13312;                        // 3072

  const int bl   = blockIdx.x;
  const int tid  = threadIdx.x;
  const int wave = tid >> 5, lane = tid & 31;

  for (int i = tid; i < 2048; i += 256) ylds[i] = y[bl * 2048 + i];
  for (int i = tid; i < 1024; i += 256) {
    int s = i >> 5, k = i & 31;
    float ang = 0.19634954084936207f * (float)(s * k);   // 2*pi/32
    Fr[i] = __cosf(ang);
    Fi[i] = -__sinf(ang);
  }
  __syncthreads();

  // g = Wb * y + 32*bb  (real & imag parts; 6x2 tiles each = 24 wave-jobs)
  for (int j = wave; j < 24; j += 8) {
    int which = j / 12, t = j % 12, tm = t >> 1, tn = t & 1;
    const float* W  = which ? Wbi : Wbr;
    const float* bp = which ? bbi : bbr;
    float* gout     = which ? gi : gr;
    v8f acc = {};
    acc = mm_tile(W + tm * 16 * EE, EE, ylds + tn * 16, SS, EE, acc, false);
    int n = lane & 15, mb = (lane >> 4) << 3;
#pragma unroll
    for (int v = 0; v < 8; ++v) {
      int c = tm * 16 + mb + v;
      gout[c * SS + tn * 16 + n] = acc[v] + 32.0f * bp[c];
    }
  }
  __syncthreads();

  // D = g * F (complex DFT along s): Dr = gr*Fr - gi*Fi ; Di = gr*Fi + gi*Fr
  for (int j = wave; j < 24; j += 8) {
    int which = j / 12, t = j % 12, tm = t >> 1, tn = t & 1;
    v8f acc = {};
    if (which == 0) {
      acc = mm_tile(gr + tm * 16 * SS, SS, Fr + tn * 16, KK, SS, acc, false);
      acc = mm_tile(gi + tm * 16 * SS, SS, Fi + tn * 16, KK, SS, acc, true);
    } else {
      acc = mm_tile(gr + tm * 16 * SS, SS, Fi + tn * 16, KK, SS, acc, false);
      acc = mm_tile(gi + tm * 16 * SS, SS, Fr + tn * 16, KK, SS, acc, false);
    }
    float* Dout = which ? Di : Dr;
    int n = lane & 15, mb = (lane >> 4) << 3;
#pragma unroll
    for (int v = 0; v < 8; ++v)
      Dout[(tm * 16 + mb + v) * KK + tn * 16 + n] = acc[v];
  }
  __syncthreads();

  // v = gamma ⊙ (Wp1 * D + bp1)  (complex), written straight to global
  for (int j = wave; j < 24; j += 8) {
    int which = j / 12, t = j % 12, tm = t >> 1, tn = t & 1;
    v8f acc = {};
    if (which == 0) {
      acc = mm_tile(Wp1r + tm * 16 * CC, CC, Dr + tn * 16, KK, CC, acc, false);
      acc = mm_tile(Wp1i + tm * 16 * CC, CC, Di + tn * 16, KK, CC, acc, true);
    } else {
      acc = mm_tile(Wp1r + tm * 16 * CC, CC, Di + tn * 16, KK, CC, acc, false);
      acc = mm_tile(Wp1i + tm * 16 * CC, CC, Dr + tn * 16, KK, CC, acc, false);
    }
    const float* bp = which ? bp1i : bp1r;
    float* vout     = which ? vi : vr;
    int n = lane & 15, mb = (lane >> 4) << 3;
#pragma unroll
    for (int v = 0; v < 8; ++v) {
      int c = tm * 16 + mb + v;
      int k = tn * 16 + n;
      float gam = __expf(plog[(2 * CC + c) * 32 + k]);
      vout[(bl * CC + c) * KK + k] = gam * (acc[v] + bp[c]);
    }
  }
}

// ---------------------------------------------------------------------------
// K3: masked LRU prefix scan over L per (b,c,k); lambda = exp(-e^nu_log + i e^th_log)
// ---------------------------------------------------------------------------
__global__ void k3_scan(const float* __restrict__ vinr, const float* __restrict__ vini,
                        const float* __restrict__ mask, const float* __restrict__ plog,
                        float* __restrict__ voutr, float* __restrict__ vouti) {
  int t = blockIdx.x * 256 + threadIdx.x;      // B*C*K = 12288
  if (t >= BB * CC * KK) return;
  int k = t & 31;
  int c = (t >> 5) % CC;
  int b = t / (CC * KK);
  float nu = __expf(plog[c * 32 + k]);
  float th = __expf(plog[(CC + c) * 32 + k]);
  float er = __expf(-nu);
  float lr = er * __cosf(th), li = er * __sinf(th);
  float ar = 0.f, ai = 0.f;
  for (int l = 0; l < LL; ++l) {
    float m = (l > 0) ? mask[b * LL + l - 1] : 0.f;   // gate the carry
    float pr = m * ar, pi = m * ai;
    int idx = ((b * LL + l) * CC + c) * KK + k;
    ar = lr * pr - li * pi + vinr[idx];
    ai = lr * pi + li * pr + vini[idx];
    voutr[idx] = ar;
    vouti[idx] = ai;
  }
}

// ---------------------------------------------------------------------------
// K4 (per (b,l) block): f = v*iF/1024 ; p = Wp2*f + bp2 ; r = Re(Wc*p + bc) ;
//                       LN stats over (e,s)
// ---------------------------------------------------------------------------
__global__ void __launch_bounds__(256)
k4_back(const float* __restrict__ vr, const float* __restrict__ vi,
        const float* __restrict__ Wp2r, const float* __restrict__ Wp2i,
        const float* __restrict__ bp2r, const float* __restrict__ bp2i,
        const float* __restrict__ Wcr, const float* __restrict__ Wci,
        const float* __restrict__ bcr,
        float* __restrict__ rws, float* __restrict__ stats) {
  __shared__ float lds[16384];                      // 64 KB
  float* Avr = lds;                                 // 3072: v_r, then p_r
  float* Avi = lds + 3072;                          // 3072: v_i, then p_i
  float* fr  = lds + 6144;                          // 3072
  float* fi  = lds + 9216;                          // 3072
  float* iFr = lds + 12288;                         // 1024 (later: reduce scratch)
  float* iFi = lds + 13312;                         // 1024
  float* rbf = lds + 14336;                         // 2048 (E x S)

  const int bl   = blockIdx.x;
  const int tid  = threadIdx.x;
  const int wave = tid >> 5, lane = tid & 31;

  for (int i = tid; i < 3072; i += 256) {
    Avr[i] = vr[bl * 3072 + i];
    Avi[i] = vi[bl * 3072 + i];
  }
  for (int i = tid; i < 1024; i += 256) {
    int k = i >> 5, s = i & 31;
    float ang = 0.19634954084936207f * (float)(k * s);
    iFr[i] = __cosf(ang) * (1.0f / 1024.0f);        // ifft2 normalization
    iFi[i] = __sinf(ang) * (1.0f / 1024.0f);
  }
  __syncthreads();

  // f = v * iF (complex inverse DFT along k)
  for (int j = wave; j < 24; j += 8) {
    int which = j / 12, t = j % 12, tm = t >> 1, tn = t & 1;
    v8f acc = {};
    if (which == 0) {
      acc = mm_tile(Avr + tm * 16 * KK, KK, iFr + tn * 16, SS, KK, acc, false);
      acc = mm_tile(Avi + tm * 16 * KK, KK, iFi + tn * 16, SS, KK, acc, true);
    } else {
      acc = mm_tile(Avr + tm * 16 * KK, KK, iFi + tn * 16, SS, KK, acc, false);
      acc = mm_tile(Avi + tm * 16 * KK, KK, iFr + tn * 16, SS, KK, acc, false);
    }
    float* fout = which ? fi : fr;
    int n = lane & 15, mb = (lane >> 4) << 3;
#pragma unroll
    for (int v = 0; v < 8; ++v)
      fout[(tm * 16 + mb + v) * SS + tn * 16 + n] = acc[v];
  }
  __syncthreads();

  // p = Wp2 * f + bp2 (complex) -> overwrite Av buffers
  for (int j = wave; j < 24; j += 8) {
    int which = j / 12, t = j % 12, tm = t >> 1, tn = t & 1;
    v8f acc = {};
    if (which == 0) {
      acc = mm_tile(Wp2r + tm * 16 * CC, CC, fr + tn * 16, SS, CC, acc, false);
      acc = mm_tile(Wp2i + tm * 16 * CC, CC, fi + tn * 16, SS, CC, acc, true);
    } else {
      acc = mm_tile(Wp2r + tm * 16 * CC, CC, fi + tn * 16, SS, CC, acc, false);
      acc = mm_tile(Wp2i + tm * 16 * CC, CC, fr + tn * 16, SS, CC, acc, false);
    }
    const float* bp = which ? bp2i : bp2r;
    float* pout     = which ? Avi : Avr;
    int n = lane & 15, mb = (lane >> 4) << 3;
#pragma unroll
    for (int v = 0; v < 8; ++v) {
      int c = tm * 16 + mb + v;
      pout[c * SS + tn * 16 + n] = acc[v] + bp[c];
    }
  }
  __syncthreads();

  // r = Re(Wc * p + bc)  (E x S, 4x2 tiles = 8 wave-jobs)
  for (int j = wave; j < 8; j += 8) {
    // all 8 waves take exactly one tile
  }
  {
    int j = wave;                                  // 8 tiles, one per wave
    int tm = j >> 1, tn = j & 1;
    v8f acc = {};
    acc = mm_tile(Wcr + tm * 16 * CC, CC, Avr + tn * 16, SS, CC, acc, false);
    acc = mm_tile(Wci + tm * 16 * CC, CC, Avi + tn * 16, SS, CC, acc, true);
    int n = lane & 15, mb = (lane >> 4) << 3;
#pragma unroll
    for (int v = 0; v < 8; ++v) {
      int e = tm * 16 + mb + v;
      float rv = acc[v] + bcr[e];
      rbf[e * SS + tn * 16 + n] = rv;
      rws[bl * 2048 + e * SS + tn * 16 + n] = rv;
    }
  }
  __syncthreads();

  // LayerNorm stats over (e,s): each s value occurs 32x in (h,w) -> plain mean/var
  float s1 = 0.f, s2 = 0.f;
  for (int i = tid; i < 2048; i += 256) {
    float tv = rbf[i];
    s1 += tv;
    s2 += tv * tv;
  }
  float* red = iFr;                                // reuse 512 floats of twiddle space
  red[tid] = s1;
  red[256 + tid] = s2;
  __syncthreads();
  for (int off = 128; off > 0; off >>= 1) {
    if (tid < off) {
      red[tid] += red[tid + off];
      red[256 + tid] += red[256 + tid + off];
    }
    __syncthreads();
  }
  if (tid == 0) {
    float mean = red[0] * (1.0f / 2048.0f);
    float var  = red[256] * (1.0f / 2048.0f) - mean * mean;
    stats[bl * 2]     = mean;
    stats[bl * 2 + 1] = rsqrtf(var + 1e-5f);
  }
}

// ---------------------------------------------------------------------------
// K5: out[bl,e,h,w] = (r[bl,e,(h+w)&31]-mean)*invstd*ln_w + ln_b + x   (float4 stream)
// ---------------------------------------------------------------------------
__global__ void __launch_bounds__(256)
k5_final(const float* __restrict__ x, const float* __restrict__ rws,
         const float* __restrict__ stats,
         const float* __restrict__ lnw, const float* __restrict__ lnb,
         float* __restrict__ out) {
  __shared__ float rl[8 * 32];
  __shared__ float st[2];
  const int bl = blockIdx.y;
  const int e0 = blockIdx.x * 8;
  const int tid = threadIdx.x;
  for (int i = tid; i < 256; i += 256) rl[i] = rws[bl * 2048 + e0 * 32 + i];
  if (tid < 2) st[tid] = stats[bl * 2 + tid];
  __syncthreads();
  const float mean = st[0], inv = st[1];
#pragma unroll
  for (int it = 0; it < 8; ++it) {
    int flat = it * 256 + tid;                 // float4 index over [8e][32h][8w4]
    int el  = flat >> 8;
    int rem = flat & 255;
    int h   = rem >> 3;
    int w4  = (rem & 7) << 2;
    int gidx = ((bl * EE + e0 + el) * HH + h) * WW + w4;
    int lidx = ((e0 + el) * HH + h) * WW + w4;
    const float4 xv = *(const float4*)(x + gidx);
    const float4 wv = *(const float4*)(lnw + lidx);
    const float4 bv = *(const float4*)(lnb + lidx);
    const float* rrow = rl + el * 32;
    float4 o;
    o.x = (rrow[(h + w4 + 0) & 31] - mean) * inv * wv.x + bv.x + xv.x;
    o.y = (rrow[(h + w4 + 1) & 31] - mean) * inv * wv.y + bv.y + xv.y;
    o.z = (rrow[(h + w4 + 2) & 31] - mean) * inv * wv.z + bv.z + xv.z;
    o.w = (rrow[(h + w4 + 3) & 31] - mean) * inv * wv.w + bv.w + xv.w;
    *(float4*)(out + gidx) = o;
  }
}

// ---------------------------------------------------------------------------
extern "C" void kernel_launch(void* const* d_in, const int* in_sizes, int n_in,
                              void* d_out, int out_size, void* d_ws, size_t ws_size,
                              hipStream_t stream) {
  const float* x    = (const float*)d_in[0];
  const float* mask = (const float*)d_in[1];
  const float* plog = (const float*)d_in[2];
  const float* Wbr  = (const float*)d_in[3];
  const float* Wbi  = (const float*)d_in[4];
  const float* bbr  = (const float*)d_in[5];
  const float* bbi  = (const float*)d_in[6];
  const float* Wp1r = (const float*)d_in[7];
  const float* Wp1i = (const float*)d_in[8];
  const float* bp1r = (const float*)d_in[9];
  const float* bp1i = (const float*)d_in[10];
  const float* Wp2r = (const float*)d_in[11];
  const float* Wp2i = (const float*)d_in[12];
  const float* bp2r = (const float*)d_in[13];
  const float* bp2i = (const float*)d_in[14];
  const float* Wcr  = (const float*)d_in[15];
  const float* Wci  = (const float*)d_in[16];
  const float* bcr  = (const float*)d_in[17];
  // d_in[18] = bc_i unused (only real part of final projection matters)
  const float* lnw  = (const float*)d_in[19];
  const float* lnb  = (const float*)d_in[20];

  float* ws    = (float*)d_ws;
  float* y     = ws;                       // BL*E*S        = 262144
  float* vinr  = y     + 262144;           // BL*C*K        = 393216
  float* vini  = vinr  + 393216;
  float* voutr = vini  + 393216;
  float* vouti = voutr + 393216;
  float* rws   = vouti + 393216;           // BL*E*S        = 262144
  float* stats = rws   + 262144;           // BL*2

  k1_diag<<<1024, 256, 0, stream>>>(x, y);
  k2_front<<<BL, 256, 0, stream>>>(y, Wbr, Wbi, bbr, bbi, Wp1r, Wp1i, bp1r, bp1i,
                                   plog, vinr, vini);
  k3_scan<<<48, 256, 0, stream>>>(vinr, vini, mask, plog, voutr, vouti);
  k4_back<<<BL, 256, 0, stream>>>(voutr, vouti, Wp2r, Wp2i, bp2r, bp2i, Wcr, Wci,
                                  bcr, rws, stats);
  dim3 g5(EE / 8, BL);
  k5_final<<<g5, 256, 0, stream>>>(x, rws, stats, lnw, lnb, (float*)d_out);
}